// SimpleGCDEC_4337916969117
// MI455X (gfx1250) — compile-verified
//
#include <hip/hip_runtime.h>

typedef __attribute__((ext_vector_type(2))) float v2f;
typedef __attribute__((ext_vector_type(8))) float v8f;

#define NROWS 10000
#define NFEAT 3000
#define NHID  32
#define KCLUS 10
#define ALPHA_F 0.2f

// ---------------------------------------------------------------------------
// C[M x 32] = A[M x LDA] * B[LDA x 32]  (+ optional bias broadcast over rows)
// One wave per 16-row tile; two 16x16 f32 accumulators cover the 32 columns.
// V_WMMA_F32_16X16X4_F32 fragment layout (wave32):
//   A (16x4):  lane l -> row m = l&15, K-half kh = l>>4; VGPR j holds A[m][2*kh+j]
//   B (4x16):  lane l -> col n = l&15, K-half kh = l>>4; VGPR j holds B[2*kh+j][n]
//   C/D 16x16: VGPR v, lane l -> row m = v + 8*(l>>4), col n = l&15
// ---------------------------------------------------------------------------
template <int LDA, bool ADD_BIAS>
__global__ void __launch_bounds__(128)
wmma_gemm_n32(const float* __restrict__ A, const float* __restrict__ B,
              const float* __restrict__ bias, float* __restrict__ C, int ntiles)
{
    const int wid  = threadIdx.x >> 5;
    const int lane = threadIdx.x & 31;
    const int tile = blockIdx.x * 4 + wid;
    if (tile >= ntiles) return;          // wave-uniform: EXEC stays all-1 for WMMA

    const int m0   = tile * 16;
    const int mcol = lane & 15;          // A row within tile; also B/C column
    const int kh   = lane >> 4;          // K-half selector

    const float* Arow = A + (size_t)(m0 + mcol) * LDA + 2 * kh;
    const float* B0   = B + mcol;        // N-tile 0 column
    const float* B1   = B + mcol + 16;   // N-tile 1 column

    v8f acc0 = {};
    v8f acc1 = {};

#pragma unroll 4
    for (int k = 0; k < LDA; k += 4) {
        // A fragment: contiguous pair -> single 8B load
        v2f a = *(const v2f*)(Arow + k);
        const int kb = k + 2 * kh;
        v2f b0, b1;
        b0.x = B0[(size_t)kb * NHID];
        b0.y = B0[(size_t)(kb + 1) * NHID];
        b1.x = B1[(size_t)kb * NHID];
        b1.y = B1[(size_t)(kb + 1) * NHID];
        acc0 = __builtin_amdgcn_wmma_f32_16x16x4_f32(
                   false, a, false, b0, (short)0, acc0, false, false);
        acc1 = __builtin_amdgcn_wmma_f32_16x16x4_f32(
                   false, a, false, b1, (short)0, acc1, false, false);
    }

    float bv0 = 0.0f, bv1 = 0.0f;
    if (ADD_BIAS) {
        bv0 = bias[mcol];
        bv1 = bias[mcol + 16];
    }

    // Store: acc[v] holds row m0 + v + 8*kh, col mcol (tile0) / mcol+16 (tile1)
    float* Crow = C + (size_t)(m0 + 8 * kh) * NHID + mcol;
#pragma unroll
    for (int v = 0; v < 8; ++v) {
        Crow[(size_t)v * NHID]      = acc0[v] + bv0;
        Crow[(size_t)v * NHID + 16] = acc1[v] + bv1;
    }
}

// ---------------------------------------------------------------------------
// Student-t soft assignment: one thread per row of z.
// d2 = |z|^2 - 2 z.mu + |mu|^2 ; q = (1/(1+d2/a+1e-8))^(a+1)/2 ; row-normalize
// ---------------------------------------------------------------------------
__global__ void __launch_bounds__(256)
soft_assign(const float* __restrict__ z, const float* __restrict__ mu,
            float* __restrict__ q)
{
    const int row = blockIdx.x * blockDim.x + threadIdx.x;
    if (row >= NROWS) return;

    const float* zp = z + (size_t)row * NHID;
    float zr[NHID];
#pragma unroll
    for (int j = 0; j < NHID; ++j) zr[j] = zp[j];

    float z2 = 0.0f;
#pragma unroll
    for (int j = 0; j < NHID; ++j) z2 += zr[j] * zr[j];

    float qv[KCLUS];
    float s = 0.0f;
#pragma unroll
    for (int c = 0; c < KCLUS; ++c) {
        const float* m = mu + c * NHID;
        float dot = 0.0f, mu2 = 0.0f;
#pragma unroll
        for (int j = 0; j < NHID; ++j) {
            float mv = m[j];
            dot += zr[j] * mv;
            mu2 += mv * mv;
        }
        float d2 = z2 - 2.0f * dot + mu2;
        float t  = 1.0f / (1.0f + d2 / ALPHA_F + 1e-8f);
        t = __powf(t, ALPHA_F + 1.0f) * 0.5f;
        qv[c] = t;
        s += t;
    }
    const float inv = 1.0f / s;
    float* qp = q + (size_t)row * KCLUS;
#pragma unroll
    for (int c = 0; c < KCLUS; ++c) qp[c] = qv[c] * inv;
}

// ---------------------------------------------------------------------------
extern "C" void kernel_launch(void* const* d_in, const int* in_sizes, int n_in,
                              void* d_out, int out_size, void* d_ws, size_t ws_size,
                              hipStream_t stream)
{
    const float* x   = (const float*)d_in[0];  // [10000, 3000]
    const float* adj = (const float*)d_in[1];  // [10000, 10000]
    const float* W   = (const float*)d_in[2];  // [3000, 32]
    const float* b   = (const float*)d_in[3];  // [32]
    const float* mu  = (const float*)d_in[4];  // [10, 32]

    float* z = (float*)d_out;                  // [10000, 32]
    float* q = z + (size_t)NROWS * NHID;       // [10000, 10]
    float* support = (float*)d_ws;             // [10000, 32] scratch

    const int ntiles = NROWS / 16;             // 625 exactly, no tail rows
    dim3 blk(128);                             // 4 waves / block
    dim3 grd((ntiles + 3) / 4);                // 157 blocks

    // support = x @ W
    wmma_gemm_n32<NFEAT, false><<<grd, blk, 0, stream>>>(x, W, nullptr, support, ntiles);
    // z = adj @ support + b
    wmma_gemm_n32<NROWS, true><<<grd, blk, 0, stream>>>(adj, support, b, z, ntiles);
    // q = student-t soft assignment
    dim3 blk3(256), grd3((NROWS + 255) / 256);
    soft_assign<<<grd3, blk3, 0, stream>>>(z, mu, q);
}